// LSTM_fullcopy_51608327028830
// MI455X (gfx1250) — compile-verified
//
#include <hip/hip_runtime.h>
#include <hip/hip_bf16.h>
#include <math.h>

// LSTM cell, DIN = NH = 4096.
// Bandwidth-bound: ~470MB of fp32 weights @ 23.3TB/s ~= 20us. All math fp32.
// GEMV via V_WMMA_F32_16X16X4_F32: 16 rows per wave, vector broadcast across
// B columns so the C accumulator carries the K-reduction. K within each WMMA
// is remapped so every lane feeds from contiguous b128 loads:
//   lane-lo supplies physical k..k+3 (2 WMMAs), lane-hi supplies k+16..k+19.

#define NHID 4096
#define KDIM 4096

typedef __attribute__((ext_vector_type(2))) float v2f;
typedef __attribute__((ext_vector_type(4))) float v4f;
typedef __attribute__((ext_vector_type(8))) float v8f;

// blocks 0..95   : Wih tiles 0..767   (rows 0..12287 of Wih; slice 3 unused by ref)
// blocks 96..223 : Whh tiles 0..1023  (rows 0..16383 of Whh)
__launch_bounds__(256)
__global__ void lstm_gemv_wmma(const float* __restrict__ Wih,
                               const float* __restrict__ Whh,
                               const float* __restrict__ x,
                               const float* __restrict__ h,
                               float* __restrict__ zi,
                               float* __restrict__ zh)
{
    __shared__ float vlds[KDIM];

    const int  tid   = threadIdx.x;
    const bool isWih = (blockIdx.x < 96);

    // Stage the 16KB vector in LDS once per block (coalesced b128s).
    const float* vsrc = isWih ? x : h;
    #pragma unroll
    for (int i = 0; i < KDIM / (256 * 4); ++i) {
        int idx = (tid + i * 256) * 4;
        *(v4f*)(&vlds[idx]) = *(const v4f*)(vsrc + idx);
    }
    __syncthreads();

    const int wave = tid >> 5;
    const int lane = tid & 31;
    const int koff = (lane >= 16) ? 16 : 0;  // lane-hi half supplies k+16..k+31
    const int tile = blockIdx.x * 8 + wave;
    const int m0   = (isWih ? tile : (tile - 768)) << 4;   // first of 16 rows

    const float* __restrict__ Wrow =
        (isWih ? Wih : Whh) + (size_t)(m0 + (lane & 15)) * KDIM + koff;
    float* __restrict__ out = isWih ? zi : zh;

    v8f acc0 = {};
    v8f acc1 = {};

    for (int k = 0; k < KDIM; k += 32) {
        #pragma unroll
        for (int j = 0; j < 4; ++j) {
            // 16B contiguous per lane; weights are single-use -> nontemporal.
            v4f a = __builtin_nontemporal_load((const v4f*)(Wrow + k + 4 * j));
            v4f b = *(const v4f*)(&vlds[k + koff + 4 * j]);
            v2f a0 = { a[0], a[1] };
            v2f b0 = { b[0], b[1] };
            v2f a1 = { a[2], a[3] };
            v2f b1 = { b[2], b[3] };
            // WMMA0: logical K = physical {k+4j, k+4j+1, k+16+4j, k+16+4j+1}
            acc0 = __builtin_amdgcn_wmma_f32_16x16x4_f32(false, a0, false, b0,
                                                         (short)0, acc0, false, false);
            // WMMA1: logical K = physical {k+4j+2, k+4j+3, k+16+4j+2, k+16+4j+3}
            acc1 = __builtin_amdgcn_wmma_f32_16x16x4_f32(false, a1, false, b1,
                                                         (short)0, acc1, false, false);
        }
    }

    v8f acc = acc0 + acc1;

    // C/D layout: lanes 0-15 vgpr j = row m0+j, lanes 16-31 vgpr j = row m0+8+j.
    // All 16 columns identical (B is a column-broadcast), so lanes 0 and 16
    // hold the 16 dot products; they store them.
    if (lane == 0 || lane == 16) {
        float* o = out + m0 + ((lane >> 4) << 3);
        #pragma unroll
        for (int j = 0; j < 8; ++j) o[j] = acc[j];
    }
}

__global__ void lstm_pointwise(const float* __restrict__ zi,
                               const float* __restrict__ zh,
                               const float* __restrict__ c,
                               const float* __restrict__ bih,
                               const float* __restrict__ bhh,
                               float* __restrict__ out)
{
    int n = blockIdx.x * blockDim.x + threadIdx.x;
    if (n >= NHID) return;

    float pi = zi[n]            + bih[n]            + zh[n]            + bhh[n];
    float pf = zi[n + NHID]     + bih[n + NHID]     + zh[n + NHID]     + bhh[n + NHID];
    float pg = zi[n + 2*NHID]   + bih[n + 2*NHID]   + zh[n + 2*NHID]   + bhh[n + 2*NHID];
    // Reference quirk: Wio == Wih[2nh:3nh]  ->  o-gate reuses the g-slice of zi,
    // but takes slice-3 biases and slice-3 of Whh.
    float po = zi[n + 2*NHID]   + bih[n + 3*NHID]   + zh[n + 3*NHID]   + bhh[n + 3*NHID];

    float it = 1.0f / (1.0f + expf(-pi));
    float ft = 1.0f / (1.0f + expf(-pf));
    float gt = tanhf(pg);
    float ot = 1.0f / (1.0f + expf(-po));
    float ct = ft * c[n] + it * gt;
    float ht = ot * tanhf(ct);

    out[n]            = it;
    out[n +   NHID]   = ft;
    out[n + 2*NHID]   = gt;
    out[n + 3*NHID]   = ot;
    out[n + 4*NHID]   = ct;
    out[n + 5*NHID]   = ht;
}

extern "C" void kernel_launch(void* const* d_in, const int* in_sizes, int n_in,
                              void* d_out, int out_size, void* d_ws, size_t ws_size,
                              hipStream_t stream) {
    const float* x   = (const float*)d_in[0];
    const float* h   = (const float*)d_in[1];
    const float* c   = (const float*)d_in[2];
    const float* Wih = (const float*)d_in[3];
    const float* Whh = (const float*)d_in[4];
    const float* bih = (const float*)d_in[5];
    const float* bhh = (const float*)d_in[6];
    float* out = (float*)d_out;

    float* zi = (float*)d_ws;        // 3*NHID floats (i, f, g input-matvecs)
    float* zh = zi + 3 * NHID;       // 4*NHID floats (i, f, g, o hidden-matvecs)

    // 96 blocks * 8 waves = 768 tiles of 16 rows -> 12288 Wih rows (i,f,g)
    // 128 blocks * 8 waves = 1024 tiles          -> 16384 Whh rows (i,f,g,o)
    lstm_gemv_wmma<<<224, 256, 0, stream>>>(Wih, Whh, x, h, zi, zh);
    lstm_pointwise<<<(NHID + 255) / 256, 256, 0, stream>>>(zi, zh, c, bih, bhh, out);
}